// PainnMessage_23313082483620
// MI455X (gfx1250) — compile-verified
//
#include <hip/hip_runtime.h>
#include <math.h>

typedef __attribute__((ext_vector_type(2))) float v2f;
typedef __attribute__((ext_vector_type(8))) float v8f;

#define NODE_DIM   128
#define NUM_IRREPS 224
#define SPH_DIM    480
#define HIDDEN     576   // NODE_DIM + 2*NUM_IRREPS
#define NUM_BASIS  20
#define N_NODES    30000
#define N_EDGES    480000
#define E_TILE     16

// ---------------------------------------------------------------------------
// Kernel 0: seed outputs with the residual (x_scalar, x_spherical)
// ---------------------------------------------------------------------------
__global__ __launch_bounds__(256) void init_out_kernel(
    const float* __restrict__ x_scalar,
    const float* __restrict__ x_spherical,
    float* __restrict__ out)
{
    const int i = blockIdx.x * 256 + threadIdx.x;
    if (i < N_NODES * NODE_DIM)  out[i] = x_scalar[i];
    if (i < N_NODES * SPH_DIM)   out[(size_t)N_NODES * NODE_DIM + i] = x_spherical[i];
}

// ---------------------------------------------------------------------------
// Kernel 1: scalar_out = silu(x_scalar @ W1 + b1) @ W2 + b2   (per 16-node tile)
// fp32 WMMA 16x16x4. 8 waves: GEMM1 -> one 16-col tile each (covers 128 cols);
// GEMM2 -> 36 col tiles round-robin.
// ---------------------------------------------------------------------------
__global__ __launch_bounds__(256) void node_mlp_kernel(
    const float* __restrict__ x_scalar,
    const float* __restrict__ W1, const float* __restrict__ b1,
    const float* __restrict__ W2, const float* __restrict__ b2,
    float* __restrict__ scalar_out)
{
    __shared__ float x_lds[E_TILE * NODE_DIM];
    __shared__ float h_lds[E_TILE * NODE_DIM];

    const int tid  = threadIdx.x;
    const int wave = tid >> 5;
    const int lane = tid & 31;
    const int node_base = blockIdx.x * E_TILE;

    // stage 16x128 x tile (rows are contiguous in memory)
    for (int i = tid; i < E_TILE * NODE_DIM; i += 256)
        x_lds[i] = x_scalar[(size_t)node_base * NODE_DIM + i];
    __syncthreads();

    const int nlo   = lane & 15;        // column within 16-wide tile
    const int khalf = (lane >> 4) * 2;  // lanes 16-31 hold K=2,3
    const int row0  = (lane >> 4) * 8;  // C layout: hi lanes hold M=8..15

    // ---- GEMM1: h = silu(x @ W1 + b1), wave w owns cols [16w, 16w+16) ----
    {
        const int n = wave * 16 + nlo;
        v8f acc = {};
        for (int k0 = 0; k0 < NODE_DIM; k0 += 4) {
            v2f a = *(const v2f*)&x_lds[nlo * NODE_DIM + k0 + khalf]; // A: M=lane%16
            v2f b;
            b.x = W1[(k0 + khalf)     * NODE_DIM + n];
            b.y = W1[(k0 + khalf + 1) * NODE_DIM + n];
            acc = __builtin_amdgcn_wmma_f32_16x16x4_f32(
                false, a, false, b, (short)0, acc, false, false);
        }
        const float bias = b1[n];
        for (int r = 0; r < 8; ++r) {
            float v = acc[r] + bias;
            v = v / (1.0f + __expf(-v));          // SiLU
            h_lds[(row0 + r) * NODE_DIM + n] = v;
        }
    }
    __syncthreads();

    // ---- GEMM2: out = h @ W2 + b2, 36 column tiles of 16 ----
    for (int t = wave; t < HIDDEN / 16; t += 8) {
        const int n = t * 16 + nlo;
        v8f acc = {};
        for (int k0 = 0; k0 < NODE_DIM; k0 += 4) {
            v2f a = *(const v2f*)&h_lds[nlo * NODE_DIM + k0 + khalf];
            v2f b;
            b.x = W2[(k0 + khalf)     * HIDDEN + n];
            b.y = W2[(k0 + khalf + 1) * HIDDEN + n];
            acc = __builtin_amdgcn_wmma_f32_16x16x4_f32(
                false, a, false, b, (short)0, acc, false, false);
        }
        const float bias = b2[n];
        for (int r = 0; r < 8; ++r)
            scalar_out[(size_t)(node_base + row0 + r) * HIDDEN + n] = acc[r] + bias;
    }
}

// ---------------------------------------------------------------------------
// Kernel 2: fused edge kernel, 16 edges per workgroup, 9 waves (288 threads).
// Phase 1: filter_weight tile via fp32 WMMA (K=20 -> 5 k4 steps), fuse
//          *fcut and gather-multiply scalar_out[src]; gate cols -> LDS,
//          message_scalar cols -> atomicAdd(new_scalar[dst]).
// Phase 2: irrep-expanded spherical gating, atomicAdd(new_spherical[dst]).
// ---------------------------------------------------------------------------
__global__ __launch_bounds__(288) void edge_kernel(
    const float* __restrict__ x_spherical,
    const float* __restrict__ rbf,
    const float* __restrict__ fcut,
    const float* __restrict__ rsh,
    const int*   __restrict__ edge_index,
    const float* __restrict__ Wrbf, const float* __restrict__ brbf,
    const float* __restrict__ scalar_out,
    float* __restrict__ new_scalar,
    float* __restrict__ new_spherical)
{
    __shared__ float rbf_lds[E_TILE][24];            // 20 padded to 24 (8B-aligned pairs)
    __shared__ float fo_lds[E_TILE][2 * NUM_IRREPS]; // gate_state | gate_edge
    __shared__ int   src_lds[E_TILE];
    __shared__ int   dst_lds[E_TILE];
    __shared__ float fcut_lds[E_TILE];

    const int tid  = threadIdx.x;
    const int wave = tid >> 5;
    const int lane = tid & 31;
    const int edge_base = blockIdx.x * E_TILE;

    // ---- Phase 0: stage per-edge data ----
    for (int i = tid; i < E_TILE * NUM_BASIS; i += 288) {
        const int e = i / NUM_BASIS, k = i - e * NUM_BASIS;
        rbf_lds[e][k] = rbf[(size_t)(edge_base + e) * NUM_BASIS + k];
    }
    if (tid < E_TILE) {
        dst_lds[tid]  = edge_index[edge_base + tid];            // row 0 = dst
        src_lds[tid]  = edge_index[N_EDGES + edge_base + tid];  // row 1 = src
        fcut_lds[tid] = fcut[edge_base + tid];
    }
    __syncthreads();

    const int nlo   = lane & 15;
    const int khalf = (lane >> 4) * 2;
    const int row0  = (lane >> 4) * 8;

    // ---- Phase 1: filter GEMM + gating split (wave w owns tiles 4w..4w+3) ----
    for (int t = wave * 4; t < wave * 4 + 4; ++t) {
        const int n = t * 16 + nlo;
        v8f acc = {};
        for (int k0 = 0; k0 < NUM_BASIS; k0 += 4) {
            v2f a = *(const v2f*)&rbf_lds[nlo][k0 + khalf];
            v2f b;
            b.x = Wrbf[(k0 + khalf)     * HIDDEN + n];
            b.y = Wrbf[(k0 + khalf + 1) * HIDDEN + n];
            acc = __builtin_amdgcn_wmma_f32_16x16x4_f32(
                false, a, false, b, (short)0, acc, false, false);
        }
        const float bias = brbf[n];
        for (int r = 0; r < 8; ++r) {
            const int e = row0 + r;
            const float fw = (acc[r] + bias) * fcut_lds[e];
            const float fo = scalar_out[(size_t)src_lds[e] * HIDDEN + n] * fw;
            if (n < 2 * NUM_IRREPS) {
                fo_lds[e][n] = fo;                                   // gates
            } else {
                atomicAdd(&new_scalar[(size_t)dst_lds[e] * NODE_DIM +
                                      (n - 2 * NUM_IRREPS)], fo);    // message_scalar
            }
        }
    }
    __syncthreads();

    // ---- Phase 2: spherical messages with irrep expansion ----
    for (int i = tid; i < E_TILE * SPH_DIM; i += 288) {
        const int e = i / SPH_DIM;
        const int s = i - e * SPH_DIM;
        // _REPEAT_IDX: 128 x l=0 (1), 64 x l=1 (3), 32 x l=2 (5)
        const int irr = (s < 128) ? s
                      : (s < 320) ? 128 + (s - 128) / 3
                                  : 192 + (s - 320) / 5;
        const float gs = fo_lds[e][irr];
        const float ge = fo_lds[e][NUM_IRREPS + irr];
        const float xs = x_spherical[(size_t)src_lds[e] * SPH_DIM + s];
        const float rh = rsh[(size_t)(edge_base + e) * SPH_DIM + s];
        atomicAdd(&new_spherical[(size_t)dst_lds[e] * SPH_DIM + s], xs * gs + rh * ge);
    }
}

// ---------------------------------------------------------------------------
extern "C" void kernel_launch(void* const* d_in, const int* in_sizes, int n_in,
                              void* d_out, int out_size, void* d_ws, size_t ws_size,
                              hipStream_t stream)
{
    const float* x_scalar    = (const float*)d_in[0];
    const float* x_spherical = (const float*)d_in[1];
    const float* rbf         = (const float*)d_in[2];
    const float* fcut        = (const float*)d_in[3];
    const float* rsh         = (const float*)d_in[4];
    const int*   edge_index  = (const int*)  d_in[5];
    const float* W1          = (const float*)d_in[6];
    const float* b1          = (const float*)d_in[7];
    const float* W2          = (const float*)d_in[8];
    const float* b2          = (const float*)d_in[9];
    const float* Wrbf        = (const float*)d_in[10];
    const float* brbf        = (const float*)d_in[11];

    float* out           = (float*)d_out;
    float* new_scalar    = out;
    float* new_spherical = out + (size_t)N_NODES * NODE_DIM;
    float* scalar_out    = (float*)d_ws;   // 30000*576 floats = 69.1 MB scratch

    init_out_kernel<<<(N_NODES * SPH_DIM + 255) / 256, 256, 0, stream>>>(
        x_scalar, x_spherical, out);

    node_mlp_kernel<<<N_NODES / E_TILE, 256, 0, stream>>>(
        x_scalar, W1, b1, W2, b2, scalar_out);

    edge_kernel<<<N_EDGES / E_TILE, 288, 0, stream>>>(
        x_spherical, rbf, fcut, rsh, edge_index, Wrbf, brbf,
        scalar_out, new_scalar, new_spherical);
}